// LinearAttention_49331994362400
// MI455X (gfx1250) — compile-verified
//
#include <hip/hip_runtime.h>
#include <hip/hip_bf16.h>

// Fused LinearAttention for MI455X (gfx1250), fp32 end-to-end using
// V_WMMA_F32_16X16X4_F32 for every GEMM stage (full reference precision).
//
// Pipeline:
//   1) proj_kv    : K = Wk@x, V = Wv@x            -> ws (L2-resident)
//   2) kstats     : per-row max & sum(exp) for k softmax over n
//   3) ctx_gemm   : ctx[b,h,d,e] = sum_n softmax_n(k)[d,n] * v[e,n]   (WMMA)
//   4) fused_attn : q = softmax_d(Wq@x)*scale; o = ctx^T q; y = Wo@o+bo (WMMA x3)
//                   + per-batch sum/sumsq accumulation for GroupNorm
//   5) gn_finalize: y = (y-mu)*rsqrt(var+eps)*w + b   (in-place on d_out)
//
// Round-2 changes: avoid pointer arrays so address-space inference keeps the
// GEMM operand loads on the GLOBAL path (global_load_b64, LOADcnt only)
// instead of FLAT (which also burns DScnt); float4-vectorized LDS staging and
// GroupNorm streaming pass.

#define BATCH 32
#define CDIM  128      // channels == HIDDEN
#define NDIM  4096     // 64*64 spatial
#define HEADS 4
#define DHEAD 32
#define QSCALE 0.17677669529663687f   // 32^-0.5

typedef __attribute__((ext_vector_type(2))) float v2f;
typedef __attribute__((ext_vector_type(8))) float v8f;

__device__ __forceinline__ v8f v8f_zero() {
  v8f z = {0.f,0.f,0.f,0.f,0.f,0.f,0.f,0.f};
  return z;
}

// D = A(16x4 f32) x B(4x16 f32) + C(16x16 f32), wave32
__device__ __forceinline__ v8f wmma4(v2f a, v2f b, v8f c) {
  return __builtin_amdgcn_wmma_f32_16x16x4_f32(
      /*neg_a=*/false, a, /*neg_b=*/false, b,
      /*c_mod=*/(short)0, c, /*reuse_a=*/false, /*reuse_b=*/false);
}

// ---------------------------------------------------------------------------
// One 128x128 @ 128x16 projection slice (per wave) for kernel 1.
// Called with kernel-argument pointers directly so the loads stay GLOBAL.
// ---------------------------------------------------------------------------
__device__ __forceinline__ void proj_one(const float* __restrict__ W,
                                         const float* __restrict__ xb,
                                         float* __restrict__ ob, int col,
                                         int ln, int lh) {
  v8f acc[8];
#pragma unroll
  for (int t = 0; t < 8; ++t) acc[t] = v8f_zero();
  for (int k0 = 0; k0 < CDIM; k0 += 4) {
    const int kA = k0 + lh * 2;
    v2f bf;
    bf.x = xb[(size_t)(kA + 0) * NDIM + col];
    bf.y = xb[(size_t)(kA + 1) * NDIM + col];
#pragma unroll
    for (int mt = 0; mt < 8; ++mt) {
      v2f af;
      af.x = W[(mt * 16 + ln) * CDIM + kA + 0];
      af.y = W[(mt * 16 + ln) * CDIM + kA + 1];
      acc[mt] = wmma4(af, bf, acc[mt]);
    }
  }
#pragma unroll
  for (int mt = 0; mt < 8; ++mt)
#pragma unroll
    for (int r = 0; r < 8; ++r)
      ob[(size_t)(mt * 16 + r + lh * 8) * NDIM + col] = acc[mt][r];
}

// ---------------------------------------------------------------------------
// Kernel 1: K/V projections.  grid (NDIM/128, BATCH), block 256 (8 waves).
// Wave w owns the 16-column slice [n0 + 16w, n0 + 16w + 16).
// ---------------------------------------------------------------------------
__global__ __launch_bounds__(256) void proj_kv_kernel(
    const float* __restrict__ x, const float* __restrict__ Wk,
    const float* __restrict__ Wv, float* __restrict__ kbuf,
    float* __restrict__ vbuf) {
  const int b   = blockIdx.y;
  const int n0  = blockIdx.x * 128;
  const int w   = threadIdx.x >> 5;
  const int ln  = threadIdx.x & 15;
  const int lh  = (threadIdx.x & 31) >> 4;
  const int col = n0 + w * 16 + ln;
  const size_t boff = (size_t)b * CDIM * NDIM;

  proj_one(Wk, x + boff, kbuf + boff, col, ln, lh);
  proj_one(Wv, x + boff, vbuf + boff, col, ln, lh);
}

// ---------------------------------------------------------------------------
// Kernel 2: per-row max and sum(exp) over n for the k softmax.
// grid(BATCH*CDIM) blocks, 256 threads each.
// ---------------------------------------------------------------------------
__global__ __launch_bounds__(256) void kstats_kernel(
    const float* __restrict__ kbuf, float* __restrict__ kmax,
    float* __restrict__ ksum) {
  __shared__ float red[256];
  const int row = blockIdx.x;
  const int tid = threadIdx.x;
  const float4* kr = (const float4*)(kbuf + (size_t)row * NDIM);

  float m = -1e30f;
  for (int i = tid; i < NDIM / 4; i += 256) {
    const float4 v = kr[i];
    m = fmaxf(m, fmaxf(fmaxf(v.x, v.y), fmaxf(v.z, v.w)));
  }
  red[tid] = m; __syncthreads();
  for (int s = 128; s > 0; s >>= 1) {
    if (tid < s) red[tid] = fmaxf(red[tid], red[tid + s]);
    __syncthreads();
  }
  const float rowm = red[0];
  __syncthreads();

  float sum = 0.f;
  for (int i = tid; i < NDIM / 4; i += 256) {
    const float4 v = kr[i];
    sum += expf(v.x - rowm) + expf(v.y - rowm) + expf(v.z - rowm) +
           expf(v.w - rowm);
  }
  red[tid] = sum; __syncthreads();
  for (int s = 128; s > 0; s >>= 1) {
    if (tid < s) red[tid] += red[tid + s];
    __syncthreads();
  }
  if (tid == 0) { kmax[row] = rowm; ksum[row] = red[0]; }
}

// ---------------------------------------------------------------------------
// Kernel 3: ctx[b,h,d,e] = sum_n exp(k[d,n]-max)/sum * v[e,n].
// grid(BATCH*HEADS) blocks, 256 threads (8 waves).
// Stage 128-n chunks of k (exp applied) and v in LDS via float4; each wave
// reduces its 16-n slice into 2x2 WMMA tiles; cross-wave LDS reduction.
// ---------------------------------------------------------------------------
__global__ __launch_bounds__(256) void ctx_kernel(
    const float* __restrict__ kbuf, const float* __restrict__ vbuf,
    const float* __restrict__ kmax, const float* __restrict__ ksum,
    float* __restrict__ ctx) {
  __shared__ float smem[8704];  // kt[32][136] | vt[32][136]; reused as partial[8][1024]
  float* kt = smem;             // kt[ch*136 + n]
  float* vt = smem + 4352;

  const int bh = blockIdx.x;
  const int b  = bh >> 2;
  const int h  = bh & 3;
  const int tid = threadIdx.x;
  const int w   = tid >> 5;
  const int ln  = tid & 15;
  const int lh  = (tid & 31) >> 4;
  const int chbase = b * CDIM + h * DHEAD;

  v8f acc[2][2];
  acc[0][0] = v8f_zero(); acc[0][1] = v8f_zero();
  acc[1][0] = v8f_zero(); acc[1][1] = v8f_zero();

  for (int chunk = 0; chunk < NDIM / 128; ++chunk) {
    const int n0 = chunk * 128;
    __syncthreads();
    // stage 32 rows x 128 cols of k (exp'ed) and v, float4 at a time
    for (int idx = tid; idx < 32 * 32; idx += 256) {
      const int ch = idx >> 5;          // row 0..31
      const int c4 = (idx & 31) * 4;    // col 0,4,...,124
      const float rm = kmax[chbase + ch];
      const float4 kv =
          *(const float4*)(kbuf + (size_t)(chbase + ch) * NDIM + n0 + c4);
      const float4 vv =
          *(const float4*)(vbuf + (size_t)(chbase + ch) * NDIM + n0 + c4);
      float4 ke;
      ke.x = expf(kv.x - rm); ke.y = expf(kv.y - rm);
      ke.z = expf(kv.z - rm); ke.w = expf(kv.w - rm);
      *(float4*)(kt + ch * 136 + c4) = ke;
      *(float4*)(vt + ch * 136 + c4) = vv;
    }
    __syncthreads();
#pragma unroll
    for (int kk = 0; kk < 16; kk += 4) {
      const int kl = w * 16 + kk + lh * 2;
      v2f a0, a1, b0, b1;
      a0.x = kt[ln * 136 + kl];        a0.y = kt[ln * 136 + kl + 1];
      a1.x = kt[(16 + ln) * 136 + kl]; a1.y = kt[(16 + ln) * 136 + kl + 1];
      b0.x = vt[ln * 136 + kl];        b0.y = vt[ln * 136 + kl + 1];
      b1.x = vt[(16 + ln) * 136 + kl]; b1.y = vt[(16 + ln) * 136 + kl + 1];
      acc[0][0] = wmma4(a0, b0, acc[0][0]);
      acc[0][1] = wmma4(a0, b1, acc[0][1]);
      acc[1][0] = wmma4(a1, b0, acc[1][0]);
      acc[1][1] = wmma4(a1, b1, acc[1][1]);
    }
  }
  __syncthreads();
  // dump per-wave partials (layout: partial[w][d*32+e])
#pragma unroll
  for (int dt = 0; dt < 2; ++dt)
#pragma unroll
    for (int et = 0; et < 2; ++et)
#pragma unroll
      for (int r = 0; r < 8; ++r) {
        const int d = dt * 16 + r + lh * 8;
        const int e = et * 16 + ln;
        smem[w * 1024 + d * 32 + e] = acc[dt][et][r];
      }
  __syncthreads();
  for (int o = tid; o < 1024; o += 256) {
    float s = 0.f;
#pragma unroll
    for (int ww = 0; ww < 8; ++ww) s += smem[ww * 1024 + o];
    const int d = o >> 5;
    ctx[(size_t)bh * 1024 + o] = s / ksum[chbase + d];
  }
}

// ---------------------------------------------------------------------------
// Kernel 4: fused q-proj -> head softmax -> ctx^T q -> Wo proj + bias
//           + GroupNorm sum/sumsq accumulation.
// grid (NDIM/64, BATCH), block 256 (8 waves). 64-column tiles.
// ---------------------------------------------------------------------------
__global__ __launch_bounds__(256) void fused_attn_kernel(
    const float* __restrict__ x, const float* __restrict__ Wq,
    const float* __restrict__ Wo, const float* __restrict__ bo,
    const float* __restrict__ ctx, float* __restrict__ out,
    float* __restrict__ stats) {
  __shared__ float qt[128 * 68];   // q tile / attn-out tile, stride 68
  __shared__ float ctxs[4096];     // ctx[h][d][e] for this batch
  __shared__ float red[256];

  const int b  = blockIdx.y;
  const int n0 = blockIdx.x * 64;
  const int tid = threadIdx.x;
  const int w   = tid >> 5;
  const int ln  = tid & 15;
  const int lh  = (tid & 31) >> 4;
  const float* xb = x + (size_t)b * CDIM * NDIM;

  for (int i = tid; i < 1024; i += 256)
    *(float4*)(ctxs + i * 4) = *(const float4*)(ctx + (size_t)b * 4096 + i * 4);

  // --- q projection: wave w computes output rows [16w, 16w+16) for 64 cols
  {
    v8f acc[4];
#pragma unroll
    for (int nb = 0; nb < 4; ++nb) acc[nb] = v8f_zero();
    for (int k0 = 0; k0 < CDIM; k0 += 4) {
      const int kA = k0 + lh * 2;
      v2f af;
      af.x = Wq[(w * 16 + ln) * CDIM + kA + 0];
      af.y = Wq[(w * 16 + ln) * CDIM + kA + 1];
#pragma unroll
      for (int nb = 0; nb < 4; ++nb) {
        const int col = n0 + nb * 16 + ln;
        v2f bf;
        bf.x = xb[(size_t)(kA + 0) * NDIM + col];
        bf.y = xb[(size_t)(kA + 1) * NDIM + col];
        acc[nb] = wmma4(af, bf, acc[nb]);
      }
    }
#pragma unroll
    for (int nb = 0; nb < 4; ++nb)
#pragma unroll
      for (int r = 0; r < 8; ++r)
        qt[(w * 16 + r + lh * 8) * 68 + nb * 16 + ln] = acc[nb][r];
  }
  __syncthreads();

  // --- q softmax over head dim (32) per column, times 1/sqrt(dh)
  {
    const int h = tid >> 6, nl = tid & 63;
    float m = -1e30f;
#pragma unroll
    for (int d = 0; d < DHEAD; ++d)
      m = fmaxf(m, qt[(h * DHEAD + d) * 68 + nl]);
    float s = 0.f;
#pragma unroll
    for (int d = 0; d < DHEAD; ++d)
      s += expf(qt[(h * DHEAD + d) * 68 + nl] - m);
    const float inv = QSCALE / s;
#pragma unroll
    for (int d = 0; d < DHEAD; ++d) {
      const int idx = (h * DHEAD + d) * 68 + nl;
      qt[idx] = expf(qt[idx] - m) * inv;
    }
  }
  __syncthreads();

  // --- attn out: o[e,n] = sum_d ctx[d,e] q[d,n], per head, in place in qt
  {
    const int et = w >> 2, nb = w & 3;
    v8f oacc[4];
#pragma unroll
    for (int h = 0; h < HEADS; ++h) {
      v8f acc = v8f_zero();
#pragma unroll
      for (int k0 = 0; k0 < DHEAD; k0 += 4) {
        const int kA = k0 + lh * 2;
        v2f af;  // A[M=e][K=d] = ctx[d][e]
        af.x = ctxs[h * 1024 + (kA + 0) * 32 + et * 16 + ln];
        af.y = ctxs[h * 1024 + (kA + 1) * 32 + et * 16 + ln];
        v2f bf;  // B[K=d][N=n] = q[d,n]
        bf.x = qt[(h * DHEAD + kA + 0) * 68 + nb * 16 + ln];
        bf.y = qt[(h * DHEAD + kA + 1) * 68 + nb * 16 + ln];
        acc = wmma4(af, bf, acc);
      }
      oacc[h] = acc;
    }
    __syncthreads();
#pragma unroll
    for (int h = 0; h < HEADS; ++h)
#pragma unroll
      for (int r = 0; r < 8; ++r)
        qt[(h * DHEAD + et * 16 + r + lh * 8) * 68 + nb * 16 + ln] = oacc[h][r];
  }
  __syncthreads();

  // --- output projection: y = Wo @ o + bo, write d_out, gather GN stats
  float lsum = 0.f, lsq = 0.f;
  {
    v8f acc[4];
#pragma unroll
    for (int nb = 0; nb < 4; ++nb) acc[nb] = v8f_zero();
    for (int k0 = 0; k0 < CDIM; k0 += 4) {
      const int kA = k0 + lh * 2;
      v2f af;
      af.x = Wo[(w * 16 + ln) * CDIM + kA + 0];
      af.y = Wo[(w * 16 + ln) * CDIM + kA + 1];
#pragma unroll
      for (int nb = 0; nb < 4; ++nb) {
        v2f bf;
        bf.x = qt[(kA + 0) * 68 + nb * 16 + ln];
        bf.y = qt[(kA + 1) * 68 + nb * 16 + ln];
        acc[nb] = wmma4(af, bf, acc[nb]);
      }
    }
#pragma unroll
    for (int nb = 0; nb < 4; ++nb)
#pragma unroll
      for (int r = 0; r < 8; ++r) {
        const int row = w * 16 + r + lh * 8;
        const int col = n0 + nb * 16 + ln;
        const float v = acc[nb][r] + bo[row];
        out[(size_t)b * CDIM * NDIM + (size_t)row * NDIM + col] = v;
        lsum += v;
        lsq += v * v;
      }
  }
  red[tid] = lsum; __syncthreads();
  for (int s = 128; s > 0; s >>= 1) {
    if (tid < s) red[tid] += red[tid + s];
    __syncthreads();
  }
  if (tid == 0) atomicAdd(&stats[b * 2 + 0], red[0]);
  __syncthreads();
  red[tid] = lsq; __syncthreads();
  for (int s = 128; s > 0; s >>= 1) {
    if (tid < s) red[tid] += red[tid + s];
    __syncthreads();
  }
  if (tid == 0) atomicAdd(&stats[b * 2 + 1], red[0]);
}

// ---------------------------------------------------------------------------
// Kernel 5: GroupNorm(1 group) finalize, in place on d_out (float4 stream).
// grid (32 chunks, BATCH), block 256; 16384 elems per block.
// ---------------------------------------------------------------------------
__global__ __launch_bounds__(256) void gn_kernel(
    float* __restrict__ out, const float* __restrict__ stats,
    const float* __restrict__ gnw, const float* __restrict__ gnb) {
  const int b = blockIdx.y;
  const float cnt = (float)(CDIM * NDIM);
  const float mu  = stats[b * 2 + 0] / cnt;
  const float var = stats[b * 2 + 1] / cnt - mu * mu;
  const float rstd = rsqrtf(var + 1e-5f);
  float4* o4 = (float4*)(out + (size_t)b * CDIM * NDIM +
                         (size_t)blockIdx.x * 16384);
  for (int i = threadIdx.x; i < 4096; i += 256) {
    const int c = (blockIdx.x * 16384 + i * 4) >> 12;  // / NDIM (row of 4096)
    const float sw = rstd * gnw[c];
    const float sb = gnb[c] - mu * sw;
    float4 v = o4[i];
    v.x = v.x * sw + sb; v.y = v.y * sw + sb;
    v.z = v.z * sw + sb; v.w = v.w * sw + sb;
    o4[i] = v;
  }
}

// ---------------------------------------------------------------------------
// Workspace layout (floats):
//   [0,64)        stats (sum,sumsq per batch)   -- memset each launch
//   [64,4160)     kmax
//   [4160,8256)   ksum
//   [8256,139328) ctx  [b][h][d][e]
//   [139328,+16M) kbuf [b][128][4096]
//   [+16M,+32M)   vbuf [b][128][4096]
// Total ~134.8 MB.
// ---------------------------------------------------------------------------
extern "C" void kernel_launch(void* const* d_in, const int* in_sizes, int n_in,
                              void* d_out, int out_size, void* d_ws,
                              size_t ws_size, hipStream_t stream) {
  const float* x   = (const float*)d_in[0];
  const float* Wq  = (const float*)d_in[1];
  const float* Wk  = (const float*)d_in[2];
  const float* Wv  = (const float*)d_in[3];
  const float* Wo  = (const float*)d_in[4];
  const float* bo  = (const float*)d_in[5];
  const float* gnw = (const float*)d_in[6];
  const float* gnb = (const float*)d_in[7];
  float* out = (float*)d_out;

  float* ws    = (float*)d_ws;
  float* stats = ws;
  float* kmax  = ws + 64;
  float* ksum  = ws + 64 + 4096;
  float* ctx   = ws + 64 + 4096 + 4096;
  float* kbuf  = ws + 139328;
  float* vbuf  = ws + 139328 + (size_t)BATCH * CDIM * NDIM;

  hipMemsetAsync(stats, 0, 64 * sizeof(float), stream);

  proj_kv_kernel<<<dim3(NDIM / 128, BATCH), 256, 0, stream>>>(x, Wk, Wv, kbuf,
                                                              vbuf);
  kstats_kernel<<<dim3(BATCH * CDIM), 256, 0, stream>>>(kbuf, kmax, ksum);
  ctx_kernel<<<dim3(BATCH * HEADS), 256, 0, stream>>>(kbuf, vbuf, kmax, ksum,
                                                      ctx);
  fused_attn_kernel<<<dim3(NDIM / 64, BATCH), 256, 0, stream>>>(
      x, Wq, Wo, bo, ctx, out, stats);
  gn_kernel<<<dim3(32, BATCH), 256, 0, stream>>>(out, stats, gnw, gnb);
}